// TripletLoss3_10393820856649
// MI455X (gfx1250) — compile-verified
//
#include <hip/hip_runtime.h>
#include <hip/hip_bf16.h>
#include <stdint.h>

#define BLK 256
#define CHUNK_F 1024          // floats staged per row per chunk (BLK threads * 4 floats)
#define EPS 1e-6f
#define MARGIN 0.5f

// ---- CDNA5 async global->LDS copy (ASYNCcnt path, CDNA5 ISA) ----
__device__ __forceinline__ void async_copy_b128(uint32_t lds_byte_off, const float* gaddr) {
    // vdst = VGPR holding LDS byte address, vaddr = 64-bit global address, no saddr
    asm volatile("global_load_async_to_lds_b128 %0, %1, off"
                 :: "v"(lds_byte_off), "v"(gaddr)
                 : "memory");
}

template <int N>
__device__ __forceinline__ void wait_asynccnt_le() {
    asm volatile("s_wait_asynccnt %0" :: "i"(N) : "memory");
}

__device__ __forceinline__ uint32_t lds_off(const void* p) {
    // AS3 -> generic cast places the LDS byte offset in the low 32 bits
    return (uint32_t)(uintptr_t)p;
}

template <int NCHUNK>
__global__ void __launch_bounds__(BLK)
triplet_rows_kernel(const float* __restrict__ feat,
                    const float* __restrict__ label,
                    const int*   __restrict__ idx1,
                    const int*   __restrict__ idx2,
                    float* __restrict__ ws, int D) {
    __shared__ alignas(16) float4 sA[2][BLK];
    __shared__ alignas(16) float4 s1b[2][BLK];
    __shared__ alignas(16) float4 s2b[2][BLK];
    __shared__ float red1[BLK / 32];
    __shared__ float red2[BLK / 32];

    const int i   = blockIdx.x;
    const int tid = threadIdx.x;
    const int j1  = idx1[i];
    const int j2  = idx2[i];
    const int nchunk = (NCHUNK > 0) ? NCHUNK : (D / CHUNK_F);
    const float* ra = feat + (size_t)i  * (size_t)D + (size_t)tid * 4;
    const float* r1 = feat + (size_t)j1 * (size_t)D + (size_t)tid * 4;
    const float* r2 = feat + (size_t)j2 * (size_t)D + (size_t)tid * 4;

    const uint32_t la0 = lds_off(&sA[0][tid]),  la1 = lds_off(&sA[1][tid]);
    const uint32_t l10 = lds_off(&s1b[0][tid]), l11 = lds_off(&s1b[1][tid]);
    const uint32_t l20 = lds_off(&s2b[0][tid]), l21 = lds_off(&s2b[1][tid]);

    auto issue = [&](int chunk, int buf) {
        const int off = chunk * CHUNK_F;
        async_copy_b128(buf ? la1 : la0, ra + off);
        async_copy_b128(buf ? l11 : l10, r1 + off);
        async_copy_b128(buf ? l21 : l20, r2 + off);
    };

    float acc1 = 0.0f, acc2 = 0.0f;

    issue(0, 0);
    if (nchunk > 1) issue(1, 1);

    #pragma unroll
    for (int k = 0; k < nchunk; ++k) {
        // in-order completion per wave: <=3 leaves only the next chunk outstanding
        if (k + 1 < nchunk) wait_asynccnt_le<3>();
        else                wait_asynccnt_le<0>();
        __syncthreads();                       // all waves' slices of chunk k landed

        const int b = k & 1;
        const float4 a  = sA[b][tid];
        const float4 t1 = s1b[b][tid];
        const float4 t2 = s2b[b][tid];
        float d;
        d = a.x - t1.x + EPS; acc1 = fmaf(d, d, acc1);
        d = a.y - t1.y + EPS; acc1 = fmaf(d, d, acc1);
        d = a.z - t1.z + EPS; acc1 = fmaf(d, d, acc1);
        d = a.w - t1.w + EPS; acc1 = fmaf(d, d, acc1);
        d = a.x - t2.x + EPS; acc2 = fmaf(d, d, acc2);
        d = a.y - t2.y + EPS; acc2 = fmaf(d, d, acc2);
        d = a.z - t2.z + EPS; acc2 = fmaf(d, d, acc2);
        d = a.w - t2.w + EPS; acc2 = fmaf(d, d, acc2);

        __syncthreads();                       // buffer b free for reuse
        if (k + 2 < nchunk) issue(k + 2, b);
    }

    // wave32 reduction
    for (int o = 16; o > 0; o >>= 1) {
        acc1 += __shfl_down(acc1, o, 32);
        acc2 += __shfl_down(acc2, o, 32);
    }
    const int wave = tid >> 5;
    const int lane = tid & 31;
    if (lane == 0) { red1[wave] = acc1; red2[wave] = acc2; }
    __syncthreads();

    if (tid == 0) {
        float S1 = 0.0f, S2 = 0.0f;
        #pragma unroll
        for (int w = 0; w < BLK / 32; ++w) { S1 += red1[w]; S2 += red2[w]; }
        const float li = label[i];
        const float d1 = fabsf(li - label[j1]);
        const float d2 = fabsf(li - label[j2]);
        const bool swp = (d1 >= d2);           // near = t2 if swap else t1
        const float sn = swp ? S2 : S1;
        const float sf = swp ? S1 : S2;
        ws[i] = fmaxf(sqrtf(sn) - sqrtf(sf) + MARGIN, 0.0f);
    }
}

// deterministic fixed-order final reduction (no float atomics -> replay-stable)
__global__ void __launch_bounds__(256)
reduce_sum_kernel(const float* __restrict__ ws, float* __restrict__ out, int n) {
    __shared__ float sm[256];
    float s = 0.0f;
    for (int i = threadIdx.x; i < n; i += 256) s += ws[i];
    sm[threadIdx.x] = s;
    __syncthreads();
    for (int o = 128; o > 0; o >>= 1) {
        if ((int)threadIdx.x < o) sm[threadIdx.x] += sm[threadIdx.x + o];
        __syncthreads();
    }
    if (threadIdx.x == 0) out[0] = sm[0];
}

extern "C" void kernel_launch(void* const* d_in, const int* in_sizes, int n_in,
                              void* d_out, int out_size, void* d_ws, size_t ws_size,
                              hipStream_t stream) {
    const float* feat  = (const float*)d_in[0];
    const float* label = (const float*)d_in[1];
    const int*   idx1  = (const int*)d_in[2];
    const int*   idx2  = (const int*)d_in[3];
    const int B = in_sizes[1];               // label is [B,1]
    const int D = in_sizes[0] / B;           // 4096 expected (multiple of CHUNK_F)
    float* ws = (float*)d_ws;                // B floats of scratch

    if (D == 4 * CHUNK_F) {
        // fully-unrolled 4-chunk double-buffered pipeline (D = 4096)
        triplet_rows_kernel<4><<<B, BLK, 0, stream>>>(feat, label, idx1, idx2, ws, D);
    } else {
        // generic runtime-chunk-count fallback (D multiple of CHUNK_F)
        triplet_rows_kernel<0><<<B, BLK, 0, stream>>>(feat, label, idx1, idx2, ws, D);
    }
    reduce_sum_kernel<<<1, 256, 0, stream>>>(ws, (float*)d_out, B);
}